// AggregatedSlatewiseGRU_34179349741789
// MI455X (gfx1250) — compile-verified
//
#include <hip/hip_runtime.h>

// CDNA5 / gfx1250: wave32, WMMA f32 16x16x4 (true fp32 matrix op),
// async global->LDS staging (ASYNCcnt) for the item-embedding stream.
typedef float v2f __attribute__((ext_vector_type(2)));
typedef float v8f __attribute__((ext_vector_type(8)));

#define Dm     256
#define Sm     50
#define Km     20
#define ROWS   16          // batch rows per workgroup
#define STRIDE 260         // LDS row stride (16B skew -> conflict-free ds_load_b64)

__device__ __forceinline__ float sigmoidf_(float x) {
  return 1.0f / (1.0f + __expf(-x));
}

// Issue this thread's share (2 x b128) of the 16x256 x_t tile as async copies.
// global_load_async_to_lds_b128: VDST = LDS byte address, VADDR = 64-bit global
// address, tracked with ASYNCcnt (CDNA5 ISA).
__device__ __forceinline__ void async_stage_x(const float* __restrict__ item_embs,
                                              int b0, int s, int k,
                                              float* xbuf, int tid) {
#pragma unroll
  for (int i = 0; i < 2; ++i) {
    int idx = tid + i * 512;
    int row = idx >> 6;
    int c4  = (idx & 63) << 2;
    const float* g = item_embs + ((((size_t)(b0 + row)) * Sm + s) * Km + k) * Dm + c4;
    unsigned lds_off = (unsigned)(uintptr_t)(xbuf + row * STRIDE + c4);
    unsigned long long ga = (unsigned long long)(uintptr_t)g;
    asm volatile("global_load_async_to_lds_b128 %0, %1, off"
                 :: "v"(lds_off), "v"(ga) : "memory");
  }
}

__device__ __forceinline__ void wait_async0() {
  asm volatile("s_wait_asynccnt 0x0" ::: "memory");
}

__global__ __launch_bounds__(512)
void gru_fused_kernel(const float* __restrict__ item_embs,
                      const float* __restrict__ user_embs,
                      const float* __restrict__ W_ih,
                      const float* __restrict__ b_ih,
                      const float* __restrict__ W_hh,
                      const float* __restrict__ b_hh,
                      const float* __restrict__ W_out,
                      const float* __restrict__ b_out,
                      float* __restrict__ out) {
  __shared__ __align__(16) float h_lds[ROWS * STRIDE];
  __shared__ __align__(16) float x_lds[2][ROWS * STRIDE];
  __shared__ __align__(16) float wout_lds[Dm];

  const int tid  = threadIdx.x;
  const int lane = tid & 31;
  const int wave = tid >> 5;           // 0..15 -> column tile AND out-row owner
  const int b0   = blockIdx.x * ROWS;  // batch tile base

  const int n    = lane & 15;          // N index within 16-wide tile
  const int klo  = (lane >> 4) << 1;   // K sub-offset: lanes 0-15 -> 0, 16-31 -> 2
  const int hi8  = (lane >> 4) << 3;   // C/D: upper lanes hold M = j+8
  const int d0   = wave << 4;          // column tile base within D

  // ---- kick off async stage of x(s=0,k=0) into buffer 0 ----
  async_stage_x(item_embs, b0, 0, 0, x_lds[0], tid);

  // ---- h0 = user_embs[b, 0, :] ----
  for (int idx = tid; idx < ROWS * (Dm / 4); idx += 512) {
    int row = idx >> 6;
    int c4  = (idx & 63) << 2;
    float4 v = *(const float4*)(user_embs + (size_t)(b0 + row) * Dm + c4);
    *(float4*)(h_lds + row * STRIDE + c4) = v;
  }
  if (tid < Dm) wout_lds[tid] = W_out[tid];

  // ---- per-lane bias scalars (column-dependent only) ----
  const int e = d0 + n;
  const float bih_r = b_ih[e], bih_z = b_ih[Dm + e], bih_n = b_ih[2 * Dm + e];
  const float bhh_r = b_hh[e], bhh_z = b_hh[Dm + e], bhh_n = b_hh[2 * Dm + e];
  const float bout  = b_out[0];

  // B-fragment base pointers: W[e][d] row-major, B(k=d, n=e) = W[e][d]
  const float* whh_r = W_hh + (size_t)e * Dm;
  const float* whh_z = W_hh + (size_t)(Dm + e) * Dm;
  const float* whh_n = W_hh + (size_t)(2 * Dm + e) * Dm;
  const float* wih_r = W_ih + (size_t)e * Dm;
  const float* wih_z = W_ih + (size_t)(Dm + e) * Dm;
  const float* wih_n = W_ih + (size_t)(2 * Dm + e) * Dm;

  // A-fragment row base: M = lane&15 (both lane halves), K = kk + klo
  const float* hrow = h_lds + (lane & 15) * STRIDE;

  v8f hs;                              // per-slate running sum of h_new (C layout)
  for (int j = 0; j < 8; ++j) hs[j] = 0.0f;

  wait_async0();                        // x(0,0) landed (this wave's portion)
  __syncthreads();                      // all portions visible; h0 visible

  // h_prev carried in registers in C/D layout (this wave owns these elements)
  float hreg[8];
#pragma unroll
  for (int j = 0; j < 8; ++j)
    hreg[j] = h_lds[(j + hi8) * STRIDE + d0 + n];

  int cur = 0;

  for (int s = 0; s < Sm; ++s) {
    for (int k = 0; k < Km; ++k) {
      // ---- prefetch next step's x tile into the other buffer (overlaps GEMM) ----
      const bool has_next = !(s == Sm - 1 && k == Km - 1);
      if (has_next) {
        int ns = (k == Km - 1) ? s + 1 : s;
        int nk = (k == Km - 1) ? 0 : k + 1;
        async_stage_x(item_embs, b0, ns, nk, x_lds[cur ^ 1], tid);
      }

      const float* xrow = x_lds[cur] + (lane & 15) * STRIDE;

      // ---- dual GEMM: gh = h @ W_hh^T, gx = x_t @ W_ih^T (3 gate tiles each) ----
      v8f acc_hr, acc_hz, acc_hn, acc_xr, acc_xz, acc_xn;
      for (int j = 0; j < 8; ++j) {
        acc_hr[j] = 0.f; acc_hz[j] = 0.f; acc_hn[j] = 0.f;
        acc_xr[j] = 0.f; acc_xz[j] = 0.f; acc_xn[j] = 0.f;
      }

      for (int kk = 0; kk < Dm; kk += 4) {
        v2f ah = *(const v2f*)(hrow + kk + klo);
        v2f ax = *(const v2f*)(xrow + kk + klo);
        v2f br = *(const v2f*)(whh_r + kk + klo);
        v2f bz = *(const v2f*)(whh_z + kk + klo);
        v2f bn = *(const v2f*)(whh_n + kk + klo);
        acc_hr = __builtin_amdgcn_wmma_f32_16x16x4_f32(false, ah, false, br, (short)0, acc_hr, false, false);
        acc_hz = __builtin_amdgcn_wmma_f32_16x16x4_f32(false, ah, false, bz, (short)0, acc_hz, false, false);
        acc_hn = __builtin_amdgcn_wmma_f32_16x16x4_f32(false, ah, false, bn, (short)0, acc_hn, false, false);
        v2f cr = *(const v2f*)(wih_r + kk + klo);
        v2f cz = *(const v2f*)(wih_z + kk + klo);
        v2f cn = *(const v2f*)(wih_n + kk + klo);
        acc_xr = __builtin_amdgcn_wmma_f32_16x16x4_f32(false, ax, false, cr, (short)0, acc_xr, false, false);
        acc_xz = __builtin_amdgcn_wmma_f32_16x16x4_f32(false, ax, false, cz, (short)0, acc_xz, false, false);
        acc_xn = __builtin_amdgcn_wmma_f32_16x16x4_f32(false, ax, false, cn, (short)0, acc_xn, false, false);
      }

      // ---- gates (C/D layout: vgpr j -> M = j + hi8, lane -> N = n) ----
#pragma unroll
      for (int j = 0; j < 8; ++j) {
        float r  = sigmoidf_(acc_xr[j] + bih_r + acc_hr[j] + bhh_r);
        float z  = sigmoidf_(acc_xz[j] + bih_z + acc_hz[j] + bhh_z);
        float ng = tanhf(acc_xn[j] + bih_n + r * (acc_hn[j] + bhh_n));
        float hv = (1.0f - z) * ng + z * hreg[j];
        hreg[j]  = hv;
        hs[j]   += hv;
      }
      __syncthreads();  // B2: all A-fragment reads of old h complete

#pragma unroll
      for (int j = 0; j < 8; ++j)
        h_lds[(j + hi8) * STRIDE + d0 + n] = hreg[j];
      __syncthreads();  // B3: h_new visible

      // ---- out[b, s, k] = dot(h_new[b], W_out) + b_out ; wave w owns row w ----
      {
        float partial = 0.0f;
        const float* hr = h_lds + wave * STRIDE;
        for (int c = lane; c < Dm; c += 32)
          partial += hr[c] * wout_lds[c];
        for (int off = 16; off > 0; off >>= 1)
          partial += __shfl_xor(partial, off, 32);
        if (lane == 0)
          out[((size_t)(b0 + wave) * Sm + s) * Km + k] = partial + bout;
      }

      // ---- slate boundary: carry = mean over the 20 steps ----
      if (k == Km - 1) {
        __syncthreads();  // B4: out-dot reads done
#pragma unroll
        for (int j = 0; j < 8; ++j) {
          hreg[j] = hs[j] * (1.0f / (float)Km);
          h_lds[(j + hi8) * STRIDE + d0 + n] = hreg[j];
          hs[j] = 0.0f;
        }
      }

      // ---- publish next x buffer; also fences out-dot reads vs next h write ----
      wait_async0();
      __syncthreads();  // B6
      cur ^= 1;
    }
  }
}

extern "C" void kernel_launch(void* const* d_in, const int* in_sizes, int n_in,
                              void* d_out, int out_size, void* d_ws, size_t ws_size,
                              hipStream_t stream) {
  const float* item_embs = (const float*)d_in[0];
  const float* user_embs = (const float*)d_in[1];
  const float* W_ih      = (const float*)d_in[2];
  const float* b_ih      = (const float*)d_in[3];
  const float* W_hh      = (const float*)d_in[4];
  const float* b_hh      = (const float*)d_in[5];
  const float* W_out     = (const float*)d_in[6];
  const float* b_out     = (const float*)d_in[7];
  float* out = (float*)d_out;

  int B = in_sizes[1] / Dm;           // user_embs is (B, 1, D)
  dim3 grid(B / ROWS);                // 16 workgroups, each owns 16 batch rows
  dim3 block(512);                    // 16 waves (wave32) -> 16 column tiles of D

  hipLaunchKernelGGL(gru_fused_kernel, grid, block, 0, stream,
                     item_embs, user_embs, W_ih, b_ih, W_hh, b_hh,
                     W_out, b_out, out);
}